// RegionProposalLayer_90245852824388
// MI455X (gfx1250) — compile-verified
//
#include <hip/hip_runtime.h>
#include <hip/hip_bf16.h>
#include <math.h>
#include <stdint.h>
#include <stddef.h>

// Problem constants (match reference)
#define B_DIM    16
#define N_DIM    131072
#define MAX_OUT  300
#define IOU_T    0.7f
#define NMS_THREADS 1024
#define N_TILES  (N_DIM / NMS_THREADS)   // 128

#define NEG_INF (-__builtin_inff())
#define KEY_NEGINF_HI 0x007FFFFFu   // order-preserving map of -inf

// ---------------- CDNA5 feature gating (device pass only) ----------------
#if defined(__HIP_DEVICE_COMPILE__)
  #define ASYNC_LDS 1
  #if __has_builtin(__builtin_amdgcn_global_load_async_to_lds_b128)
    #define ASYNC_BUILTIN 1
  #endif
#endif

typedef int v4i_t __attribute__((ext_vector_type(4)));

#if defined(ASYNC_LDS)
__device__ __forceinline__ void async_ld_b128(const float4* g, float4* l) {
#if defined(ASYNC_BUILTIN)
  // Verified prototype: (AS1 v4i*, AS3 v4i*, imm offset, imm cpol)
  __builtin_amdgcn_global_load_async_to_lds_b128(
      (__attribute__((address_space(1))) v4i_t*)g,
      (__attribute__((address_space(3))) v4i_t*)l, 0, 0);
#else
  unsigned lofs = (unsigned)(uintptr_t)(__attribute__((address_space(3))) void*)l;
  asm volatile("global_load_async_to_lds_b128 %0, %1, off"
               :: "v"(lofs), "v"((unsigned long long)(uintptr_t)g)
               : "memory");
#endif
}

template <int C>
__device__ __forceinline__ void wait_asynccnt_le() {
#if __has_builtin(__builtin_amdgcn_s_wait_asynccnt)
  __builtin_amdgcn_s_wait_asynccnt(C);
#else
  asm volatile("s_wait_asynccnt %0" :: "i"(C) : "memory");
#endif
}

__device__ __forceinline__ void wait_dscnt0() {
#if __has_builtin(__builtin_amdgcn_s_wait_dscnt)
  __builtin_amdgcn_s_wait_dscnt(0);
#else
  asm volatile("s_wait_dscnt 0" ::: "memory");
#endif
}
#endif  // ASYNC_LDS

// Monotone (score, index) -> u64 key. max(key) == argmax(score) with
// lowest-index tie-break (matches jnp.argmax), incl. all -inf -> index 0.
__device__ __forceinline__ unsigned long long pack_key(float s, int i) {
  unsigned u = __float_as_uint(s);
  u = (u & 0x80000000u) ? ~u : (u | 0x80000000u);
  return ((unsigned long long)u << 32) | (unsigned)(~i);
}

__device__ __forceinline__ unsigned long long wave_max_u64(unsigned long long k) {
  #pragma unroll
  for (int off = 16; off > 0; off >>= 1) {
    const unsigned long long o = __shfl_xor(k, off, 32);  // wave32 butterfly
    if (o > k) k = o;
  }
  return k;
}

// ---------------- Kernel 1: decode + clip (bandwidth bound) ----------------
__global__ __launch_bounds__(256) void rpn_decode_kernel(
    const float2* __restrict__ probs,    // [B*N] (pairs)
    const float4* __restrict__ deltas,   // [B*N]
    const float4* __restrict__ anchors,  // [N]
    float4* __restrict__ boxes,          // [B*N]  (ws)
    float* __restrict__ scores)          // [B*N]  (ws)
{
  const int gid = blockIdx.x * 256 + threadIdx.x;  // exact grid, no tail
  const int i = gid & (N_DIM - 1);
  const float4 d = deltas[gid];
  const float4 a = anchors[i];
  float4 o;
  o.x = d.x * a.z + a.x;
  o.y = d.y * a.w + a.y;
  o.z = expf(d.z) * a.z;
  o.w = expf(d.w) * a.w;
  o.x = fminf(fmaxf(o.x, 0.0f), 1333.0f);
  o.y = fminf(fmaxf(o.y, 0.0f), 1333.0f);
  o.z = fminf(fmaxf(o.z, 0.0f), 1333.0f);
  o.w = fminf(fmaxf(o.w, 0.0f), 1333.0f);
  boxes[gid] = o;
  scores[gid] = probs[gid].y;  // rpn_probs[..., 1]
}

// ---------------- Kernel 2: per-batch greedy NMS ----------------
// One 1024-thread (32-wave) workgroup per batch. 300 iterations of fused
// lazy-suppress + argmax; boxes streamed through LDS via async double buffer;
// argmax via wave32 shuffle butterflies (2 barriers/step instead of 11).
__global__ __launch_bounds__(NMS_THREADS) void rpn_nms_kernel(
    const float4* __restrict__ boxes,   // [B*N]
    float* __restrict__ scores,         // [B*N] (rebuilt by decode each launch)
    float* __restrict__ out_boxes,      // [B*MAX_OUT*4]
    int* __restrict__ out_bidx,         // [B*MAX_OUT]
    int* __restrict__ out_valid)        // [B]
{
  __shared__ unsigned long long wkeys[NMS_THREADS / 32];  // per-wave partials
  __shared__ float sbox[5];             // selected box b0..b3, area
  __shared__ int   shave;               // selected box valid for suppression?
  __shared__ int   sel[MAX_OUT];
#if defined(ASYNC_LDS)
  __shared__ float4 tbuf[2][NMS_THREADS];  // 32 KB double buffer
#endif

  const int b = blockIdx.x;
  const int tid = threadIdx.x;
  const float4* __restrict__ bx = boxes + (size_t)b * N_DIM;
  float* __restrict__ sc = scores + (size_t)b * N_DIM;

  int vcount = 0;
  if (tid == 0) shave = 0;
  __syncthreads();

  for (int t = 0; t < MAX_OUT; ++t) {
    const int have = shave;  // uniform
    float b0 = 0.f, b1 = 0.f, b2 = 0.f, b3 = 0.f, ba = 0.f;
    if (have) { b0 = sbox[0]; b1 = sbox[1]; b2 = sbox[2]; b3 = sbox[3]; ba = sbox[4]; }

    unsigned long long bestkey = 0ull;

    if (have) {
#if defined(ASYNC_LDS)
      // async double-buffered stream of box tiles into LDS
      wait_dscnt0();
      async_ld_b128(bx + tid, &tbuf[0][tid]);
      for (int tile = 0; tile < N_TILES; ++tile) {
        const int cur = tile & 1;
        const int i = tile * NMS_THREADS + tid;
        if (tile + 1 < N_TILES) {
          wait_dscnt0();
          async_ld_b128(bx + i + NMS_THREADS, &tbuf[cur ^ 1][tid]);
        }
        if (i + 2 * NMS_THREADS < N_DIM)
          __builtin_prefetch(&sc[i + 2 * NMS_THREADS], 0, 3);  // global_prefetch_b8
        float s = sc[i];
        if (tile + 1 < N_TILES) wait_asynccnt_le<1>(); else wait_asynccnt_le<0>();
        if (s > NEG_INF) {
          const float4 c = tbuf[cur][tid];
          const float iy1 = fmaxf(c.x, b0);
          const float ix1 = fmaxf(c.y, b1);
          const float iy2 = fminf(c.z, b2);
          const float ix2 = fminf(c.w, b3);
          const float inter = fmaxf(iy2 - iy1, 0.0f) * fmaxf(ix2 - ix1, 0.0f);
          const float area = (c.z - c.x) * (c.w - c.y);
          const float iou = inter / (area + ba - inter);  // 0/0 -> NaN -> no suppress
          if (iou > IOU_T) { s = NEG_INF; sc[i] = NEG_INF; }
        }
        const unsigned long long k = pack_key(s, i);
        if (k > bestkey) bestkey = k;
      }
#else
      for (int i = tid; i < N_DIM; i += NMS_THREADS) {
        float s = sc[i];
        if (s > NEG_INF) {
          const float4 c = bx[i];
          const float iy1 = fmaxf(c.x, b0);
          const float ix1 = fmaxf(c.y, b1);
          const float iy2 = fminf(c.z, b2);
          const float ix2 = fminf(c.w, b3);
          const float inter = fmaxf(iy2 - iy1, 0.0f) * fmaxf(ix2 - ix1, 0.0f);
          const float area = (c.z - c.x) * (c.w - c.y);
          const float iou = inter / (area + ba - inter);
          if (iou > IOU_T) { s = NEG_INF; sc[i] = NEG_INF; }
        }
        const unsigned long long k = pack_key(s, i);
        if (k > bestkey) bestkey = k;
      }
#endif
    } else {
      // pure argmax pass (first iteration / after exhaustion)
      for (int i = tid; i < N_DIM; i += NMS_THREADS) {
        if (i + 8 * NMS_THREADS < N_DIM)
          __builtin_prefetch(&sc[i + 8 * NMS_THREADS], 0, 3);  // global_prefetch_b8
        const unsigned long long k = pack_key(sc[i], i);
        if (k > bestkey) bestkey = k;
      }
    }

    // wave32 butterfly, then one wave over the 32 partials
    bestkey = wave_max_u64(bestkey);
    if ((tid & 31) == 0) wkeys[tid >> 5] = bestkey;
    __syncthreads();
    if (tid < 32) {
      unsigned long long k = wave_max_u64(wkeys[tid]);
      if (tid == 0) {
        const unsigned bu = (unsigned)(k >> 32);
        int idx = (int)~(unsigned)k;
        if ((unsigned)idx >= (unsigned)N_DIM) idx = 0;  // safety
        const int valid = (bu > KEY_NEGINF_HI) ? 1 : 0; // score > -inf
        sel[t] = valid ? idx : 0;
        vcount += valid;
        const float4 c = bx[idx];
        sbox[0] = c.x; sbox[1] = c.y; sbox[2] = c.z; sbox[3] = c.w;
        sbox[4] = (c.z - c.x) * (c.w - c.y);
        shave = valid;
        sc[idx] = NEG_INF;  // explicit kill (matches work.at[idx].set(-inf))
      }
    }
    __syncthreads();
  }

  // Gather outputs
  for (int t2 = tid; t2 < MAX_OUT; t2 += NMS_THREADS) {
    const float4 c = bx[sel[t2]];
    reinterpret_cast<float4*>(out_boxes)[(size_t)b * MAX_OUT + t2] = c;
    out_bidx[b * MAX_OUT + t2] = b;   // repeat(arange(B), 300)
  }
  if (tid == 0) out_valid[b] = vcount;
}

// ---------------- launch ----------------
extern "C" void kernel_launch(void* const* d_in, const int* in_sizes, int n_in,
                              void* d_out, int out_size, void* d_ws, size_t ws_size,
                              hipStream_t stream) {
  (void)in_sizes; (void)n_in; (void)out_size; (void)ws_size;
  const float2* probs   = (const float2*)d_in[0];   // [B,N,2]
  const float4* deltas  = (const float4*)d_in[1];   // [B,N,4]
  const float4* anchors = (const float4*)d_in[2];   // [N,4]

  // workspace: boxes (32 MB) then scores (8 MB)
  float4* boxes = (float4*)d_ws;
  float* scores = (float*)((char*)d_ws + sizeof(float4) * (size_t)B_DIM * N_DIM);

  // output layout: [B*300*4 f32 boxes][B*300 i32 batch idx][B i32 num_valid]
  float* out_boxes = (float*)d_out;
  int* out_bidx  = (int*)(out_boxes + (size_t)B_DIM * MAX_OUT * 4);
  int* out_valid = (int*)(out_boxes + (size_t)B_DIM * MAX_OUT * 4 + (size_t)B_DIM * MAX_OUT);

  rpn_decode_kernel<<<(B_DIM * N_DIM) / 256, 256, 0, stream>>>(
      probs, deltas, anchors, boxes, scores);
  rpn_nms_kernel<<<B_DIM, NMS_THREADS, 0, stream>>>(
      boxes, scores, out_boxes, out_bidx, out_valid);
}